// TranslationLayer_73229192396909
// MI455X (gfx1250) — compile-verified
//
#include <hip/hip_runtime.h>
#include <stdint.h>

#define WIN 256
#define NB  1024
#define ROWS_PER_BLOCK 8   // one row per wave32, 8 waves per block

typedef float float4_t __attribute__((ext_vector_type(4)));
typedef int   int4v    __attribute__((__vector_size__(4 * sizeof(int))));
typedef __attribute__((address_space(1))) int4v* gbl_int4p;   // printed as "__device__" by clang
typedef __attribute__((address_space(3))) int4v* lds_int4p;   // printed as "__shared__" by clang

// Low 32 bits of a generic pointer into LDS == LDS byte offset (ISA 10.2:
// flat LDS aperture maps addr[31:0] directly; aperture base low bits are 0).
__device__ __forceinline__ uint32_t lds_offset_of(const void* p) {
  return (uint32_t)(uintptr_t)p;
}

__device__ __forceinline__ void async_row_load_b128(const float* gsrc, float* ldst) {
#if defined(__gfx1250__) && __has_builtin(__builtin_amdgcn_global_load_async_to_lds_b128)
  __builtin_amdgcn_global_load_async_to_lds_b128(
      (gbl_int4p)(uintptr_t)gsrc,
      (lds_int4p)(uintptr_t)lds_offset_of(ldst),
      0, 0);
#else
  asm volatile("global_load_async_to_lds_b128 %0, %1, off"
               :
               : "v"(lds_offset_of(ldst)), "v"(gsrc)
               : "memory");
#endif
}

__device__ __forceinline__ void wait_async_zero() {
#if defined(__gfx1250__) && __has_builtin(__builtin_amdgcn_s_wait_asynccnt)
  __builtin_amdgcn_s_wait_asynccnt(0);
#else
  asm volatile("s_wait_asynccnt 0" ::: "memory");
#endif
  asm volatile("" ::: "memory"); // keep LDS reads below the wait
}

__global__ __launch_bounds__(256) void TranslationLayer_shift_kernel(
    const float* __restrict__ in,   // [B, WIN, WIN, 1]
    const float* __restrict__ dxv,  // [B]
    const float* __restrict__ dyv,  // [B]
    float* __restrict__ out)        // [B, WIN, WIN, 1]
{
  __shared__ float lds[ROWS_PER_BLOCK * WIN];   // 8 KiB: one 1 KiB row per wave

  const int tid  = threadIdx.x;
  const int wave = tid >> 5;
  const int lane = tid & 31;

  const int row = blockIdx.x * ROWS_PER_BLOCK + wave; // global output row id
  const int b   = blockIdx.x >> 5;                    // == row >> 8 (8 rows/block, 256 rows/batch)
  const int y   = row & (WIN - 1);

  // astype(int32) == C truncation toward zero
  const int dxi = (int)dxv[b];
  const int dyi = (int)dyv[b];
  const int src_y = y + dyi;

  float* ldsRow = &lds[wave * WIN];
  float* gdst   = out + (size_t)row * WIN;

  if (src_y >= 0 && src_y < WIN) {           // wave-uniform branch at runtime
    const float* gsrc = in + ((size_t)b * WIN + (size_t)src_y) * WIN;

    // Stage the aligned 1 KiB source row into LDS: 2 x b128 per lane, async.
    async_row_load_b128(gsrc + lane * 4,        ldsRow + lane * 4);
    async_row_load_b128(gsrc + 128 + lane * 4,  ldsRow + 128 + lane * 4);
    wait_async_zero();

    // Shifted read from LDS (consecutive lanes read LDS words 4 apart),
    // then aligned 128-bit stores to global.
#pragma unroll
    for (int h = 0; h < 2; ++h) {
      const int base_x = h * 128 + lane * 4;
      float4_t v;
#pragma unroll
      for (int j = 0; j < 4; ++j) {
        const int  sx = base_x + j - dxi;
        const bool ok = (unsigned)sx < (unsigned)WIN;
        const float val = ldsRow[ok ? sx : 0];
        v[j] = ok ? val : 0.0f;
      }
      *(float4_t*)(gdst + base_x) = v;
    }
  } else {
    const float4_t z = {0.0f, 0.0f, 0.0f, 0.0f};
    *(float4_t*)(gdst + lane * 4)       = z;
    *(float4_t*)(gdst + 128 + lane * 4) = z;
  }
}

extern "C" void kernel_launch(void* const* d_in, const int* in_sizes, int n_in,
                              void* d_out, int out_size, void* d_ws, size_t ws_size,
                              hipStream_t stream) {
  const float* in  = (const float*)d_in[0];
  const float* dxv = (const float*)d_in[1];
  const float* dyv = (const float*)d_in[2];
  float* out = (float*)d_out;

  const int total_rows = NB * WIN;                      // 262144
  dim3 grid(total_rows / ROWS_PER_BLOCK);               // 32768 blocks
  dim3 block(256);                                      // 8 wave32
  TranslationLayer_shift_kernel<<<grid, block, 0, stream>>>(in, dxv, dyv, out);
}